// LinearAttention_56487409877073
// MI455X (gfx1250) — compile-verified
//
#include <hip/hip_runtime.h>
#include <math.h>

// Problem constants (B,N,L,S,H,D,M) = (4,8,1024,1024,8,64,64)
#define BB 4
#define NN 8
#define LL 1024
#define SS 1024
#define HH 8
#define DD 64
#define MM 64
#define BNH (BB * NN * HH)   // 256
#define EPSV 1e-6f

typedef float v2f __attribute__((ext_vector_type(2)));
typedef float v8f __attribute__((ext_vector_type(8)));

// elu(x) + 1  ==  x>0 ? x+1 : exp(x)
__device__ __forceinline__ float elu1(float x) {
    return x > 0.0f ? x + 1.0f : __expf(x);
}

// -----------------------------------------------------------------------------
// Kernel 1: per (b,n,h) compute KV[m][d] = sum_s K[s,d]*V[s,m]  (64x64)
//           and Ksum[d] = sum_s K[s,d], with K = (elu(keys)+1)*mask.
// grid.x = BNH, block = 256 threads (8 waves); each wave owns two 16x16 tiles.
// -----------------------------------------------------------------------------
__global__ __launch_bounds__(256) void la_kv_kernel(
    const float* __restrict__ keys,      // [B,N,S,H,D]
    const float* __restrict__ values,    // [B,N,S,H,M]
    const float* __restrict__ key_mask,  // [B,S]
    float* __restrict__ kv_ws,           // [BNH][64][64]  (m-major)
    float* __restrict__ ksum_ws)         // [BNH][64]
{
    const int bnh = blockIdx.x;
    const int h   = bnh % HH;
    const int bn  = bnh / HH;   // b*N + n
    const int b   = bn / NN;

    const float* Kp = keys   + (size_t)bn * SS * HH * DD + (size_t)h * DD; // (s,d): Kp[s*H*D + d]
    const float* Vp = values + (size_t)bn * SS * HH * MM + (size_t)h * MM; // (s,m): Vp[s*H*M + m]
    const float* maskp = key_mask + (size_t)b * SS;

    __shared__ float Kt[32][DD + 1];
    __shared__ float Vt[32][MM + 1];
    __shared__ float ksum_part[4][DD];

    const int tid  = threadIdx.x;
    const int lane = tid & 31;
    const int wave = tid >> 5;

    // two 16x16 tiles of the 64x64 (m x d) output per wave
    const int t0  = wave * 2;
    const int t1  = wave * 2 + 1;
    const int mt0 = (t0 >> 2) * 16, dt0 = (t0 & 3) * 16;
    const int mt1 = (t1 >> 2) * 16, dt1 = (t1 & 3) * 16;
    v8f acc0 = {};
    v8f acc1 = {};

    // Ksum partials: thread owns column d = tid&63, s-subgroup = tid>>6
    const int kcol = tid & 63;
    const int ksg  = tid >> 6;
    float ksum_acc = 0.0f;

    for (int s0 = 0; s0 < SS; s0 += 32) {
        // cooperative staged load: 32 rows x 64 cols (float4), fused feature map on K
        for (int i = tid; i < 512; i += 256) {
            const int row = i >> 4;
            const int c4  = (i & 15) * 4;
            const int s   = s0 + row;
            const float msk = maskp[s];
            const float4 k4 = *(const float4*)(Kp + (size_t)s * (HH * DD) + c4);
            Kt[row][c4 + 0] = elu1(k4.x) * msk;
            Kt[row][c4 + 1] = elu1(k4.y) * msk;
            Kt[row][c4 + 2] = elu1(k4.z) * msk;
            Kt[row][c4 + 3] = elu1(k4.w) * msk;
            const float4 v4 = *(const float4*)(Vp + (size_t)s * (HH * MM) + c4);
            Vt[row][c4 + 0] = v4.x;
            Vt[row][c4 + 1] = v4.y;
            Vt[row][c4 + 2] = v4.z;
            Vt[row][c4 + 3] = v4.w;
        }
        __syncthreads();

        // Ksum partial accumulation
        #pragma unroll
        for (int ss = 0; ss < 8; ++ss)
            ksum_acc += Kt[ksg * 8 + ss][kcol];

        // WMMA: KV-tile += A(16m x 4s) * B(4s x 16d), 8 sub-steps of K=4
        const int lm  = lane & 15;
        const int lhi = (lane >> 4) * 2;   // A/B VGPR layout: lanes 16-31 hold K=2,3
        #pragma unroll
        for (int kk = 0; kk < 8; ++kk) {
            const int sb = kk * 4 + lhi;
            v2f a0, b0, a1, b1;
            a0.x = Vt[sb][mt0 + lm];      a0.y = Vt[sb + 1][mt0 + lm];
            b0.x = Kt[sb][dt0 + lm];      b0.y = Kt[sb + 1][dt0 + lm];
            acc0 = __builtin_amdgcn_wmma_f32_16x16x4_f32(
                       false, a0, false, b0, (short)0, acc0, false, false);
            a1.x = Vt[sb][mt1 + lm];      a1.y = Vt[sb + 1][mt1 + lm];
            b1.x = Kt[sb][dt1 + lm];      b1.y = Kt[sb + 1][dt1 + lm];
            acc1 = __builtin_amdgcn_wmma_f32_16x16x4_f32(
                       false, a1, false, b1, (short)0, acc1, false, false);
        }
        __syncthreads();
    }

    // reduce Ksum
    ksum_part[ksg][kcol] = ksum_acc;
    __syncthreads();
    if (tid < DD) {
        ksum_ws[(size_t)bnh * DD + tid] =
            ksum_part[0][tid] + ksum_part[1][tid] + ksum_part[2][tid] + ksum_part[3][tid];
    }

    // store KV tiles.  C layout: VGPR j -> row M=j (lanes 0-15) / M=j+8 (lanes 16-31)
    float* kvp = kv_ws + (size_t)bnh * (MM * DD);
    const int cn = lane & 15;
    const int mh = (lane >> 4) * 8;
    #pragma unroll
    for (int j = 0; j < 8; ++j) {
        kvp[(size_t)(mt0 + mh + j) * DD + (dt0 + cn)] = acc0[j];
        kvp[(size_t)(mt1 + mh + j) * DD + (dt1 + cn)] = acc1[j];
    }
}

// -----------------------------------------------------------------------------
// Kernel 2: per (b,n,h, 64-row L-tile):
//   Out[l,m] = Z[l] * sum_d Q[l,d] * KV[m,d],  Z[l] = 1/(Q[l,:].Ksum + eps)
// grid = (L/64, BNH), block = 256 (8 waves, 2 output tiles each).
// -----------------------------------------------------------------------------
__global__ __launch_bounds__(256) void la_out_kernel(
    const float* __restrict__ queries,   // [B,N,L,H,D]
    const float* __restrict__ kv_ws,     // [BNH][64][64]
    const float* __restrict__ ksum_ws,   // [BNH][64]
    float* __restrict__ out)             // [B,N,L,H,M]
{
    const int lt  = blockIdx.x;   // 0..15 (64 L-rows each)
    const int bnh = blockIdx.y;   // 0..255
    const int h   = bnh % HH;
    const int bn  = bnh / HH;

    const float* Qp = queries + (size_t)bn * LL * HH * DD
                              + (size_t)(lt * 64) * (HH * DD) + (size_t)h * DD;

    __shared__ float Qt[64][DD + 1];
    __shared__ float KVs[MM][DD + 1];
    __shared__ float ksum_s[DD];
    __shared__ float zbuf[64];

    const int tid  = threadIdx.x;
    const int lane = tid & 31;
    const int wave = tid >> 5;

    // stage elu(Q)+1 tile: 64x64 floats
    for (int i = tid; i < 1024; i += 256) {
        const int row = i >> 4;
        const int c4  = (i & 15) * 4;
        const float4 q4 = *(const float4*)(Qp + (size_t)row * (HH * DD) + c4);
        Qt[row][c4 + 0] = elu1(q4.x);
        Qt[row][c4 + 1] = elu1(q4.y);
        Qt[row][c4 + 2] = elu1(q4.z);
        Qt[row][c4 + 3] = elu1(q4.w);
    }
    // stage KV (hot in L2 from kernel 1)
    const float* kvp = kv_ws + (size_t)bnh * (MM * DD);
    for (int i = tid; i < 1024; i += 256) {
        const int row = i >> 4;
        const int c4  = (i & 15) * 4;
        const float4 v4 = *(const float4*)(kvp + (size_t)row * DD + c4);
        KVs[row][c4 + 0] = v4.x;
        KVs[row][c4 + 1] = v4.y;
        KVs[row][c4 + 2] = v4.z;
        KVs[row][c4 + 3] = v4.w;
    }
    if (tid < DD) ksum_s[tid] = ksum_ws[(size_t)bnh * DD + tid];
    __syncthreads();

    // per-row normalizer
    if (tid < 64) {
        float acc = 0.0f;
        #pragma unroll
        for (int d = 0; d < DD; ++d) acc += Qt[tid][d] * ksum_s[d];
        zbuf[tid] = 1.0f / (acc + EPSV);
    }
    __syncthreads();

    // GEMM: Out-tile(16l x 16m) += A(16l x 4d) * B(4d x 16m), 16 steps over D=64
    const int t0 = wave * 2;
    const int t1 = wave * 2 + 1;
    const int l0 = (t0 >> 2) * 16, m0 = (t0 & 3) * 16;
    const int l1 = (t1 >> 2) * 16, m1 = (t1 & 3) * 16;
    v8f acc0 = {};
    v8f acc1 = {};
    const int ln  = lane & 15;
    const int lhi = (lane >> 4) * 2;
    #pragma unroll
    for (int kd = 0; kd < 16; ++kd) {
        const int db = kd * 4 + lhi;
        v2f a0, b0, a1, b1;
        a0.x = Qt[l0 + ln][db];        a0.y = Qt[l0 + ln][db + 1];
        b0.x = KVs[m0 + ln][db];       b0.y = KVs[m0 + ln][db + 1];   // B[k][n] = KV[m][d]
        acc0 = __builtin_amdgcn_wmma_f32_16x16x4_f32(
                   false, a0, false, b0, (short)0, acc0, false, false);
        a1.x = Qt[l1 + ln][db];        a1.y = Qt[l1 + ln][db + 1];
        b1.x = KVs[m1 + ln][db];       b1.y = KVs[m1 + ln][db + 1];
        acc1 = __builtin_amdgcn_wmma_f32_16x16x4_f32(
                   false, a1, false, b1, (short)0, acc1, false, false);
    }

    // scale by Z and store: out[(bn*L + lt*64 + l)*H*M + h*M + m]
    float* outp = out + (size_t)bn * LL * HH * MM
                      + (size_t)(lt * 64) * (HH * MM) + (size_t)h * MM;
    const int mh = (lane >> 4) * 8;
    #pragma unroll
    for (int j = 0; j < 8; ++j) {
        const int lr0 = l0 + mh + j;
        outp[(size_t)lr0 * (HH * MM) + (m0 + ln)] = acc0[j] * zbuf[lr0];
        const int lr1 = l1 + mh + j;
        outp[(size_t)lr1 * (HH * MM) + (m1 + ln)] = acc1[j] * zbuf[lr1];
    }
}

extern "C" void kernel_launch(void* const* d_in, const int* in_sizes, int n_in,
                              void* d_out, int out_size, void* d_ws, size_t ws_size,
                              hipStream_t stream) {
    const float* queries  = (const float*)d_in[0];
    const float* keys     = (const float*)d_in[1];
    const float* values   = (const float*)d_in[2];
    const float* key_mask = (const float*)d_in[3];
    float* out = (float*)d_out;

    float* kv_ws   = (float*)d_ws;                      // BNH * 64 * 64 floats (4 MiB)
    float* ksum_ws = kv_ws + (size_t)BNH * MM * DD;     // BNH * 64 floats

    la_kv_kernel<<<dim3(BNH), dim3(256), 0, stream>>>(keys, values, key_mask, kv_ws, ksum_ws);
    la_out_kernel<<<dim3(LL / 64, BNH), dim3(256), 0, stream>>>(queries, kv_ws, ksum_ws, out);
}